// HGTEncoder_24180665876640
// MI455X (gfx1250) — compile-verified
//
#include <hip/hip_runtime.h>
#include <math.h>

// ---------------- problem constants ----------------
#define HIDDEN 128
#define HEADS  8
#define HD     16
#define ND     10000
#define NP     20000
#define NTOT   30000
#define NEDGE  150000
#define ATT_SCALE 0.25f   // 1/sqrt(16)

typedef float v2f __attribute__((ext_vector_type(2)));
typedef float v8f __attribute__((ext_vector_type(8)));

// Only check on the DEVICE pass: amdgcn builtins are not visible to
// __has_builtin during the host (x86) compilation of HIP sources.
#if defined(__HIP_DEVICE_COMPILE__) && !__has_builtin(__builtin_amdgcn_wmma_f32_16x16x4_f32)
#error "gfx1250 wmma f32 16x16x4 builtin not available on this toolchain"
#endif

__device__ __forceinline__ v8f wmma4(v2f a, v2f b, v8f c) {
  // D = A(16x4) * B(4x16) + C, fp32 on the WMMA pipe
  return __builtin_amdgcn_wmma_f32_16x16x4_f32(false, a, false, b, (short)0, c,
                                               false, false);
}

__device__ __forceinline__ void atomicMaxF(float* addr, float v) {
  // sign-split trick: works for all finite floats with -inf init
  if (v >= 0.0f) atomicMax((int*)addr, __float_as_int(v));
  else           atomicMin((unsigned int*)addr, __float_as_uint(v));
}

// ---------------- K/Q/V projection: Y = X@W + b (3 weights fused) ----------
__global__ __launch_bounds__(256) void kqv_kernel(
    const float* __restrict__ X,
    const float* __restrict__ Wk, const float* __restrict__ bk,
    const float* __restrict__ Wq, const float* __restrict__ bq,
    const float* __restrict__ Wv, const float* __restrict__ bv,
    float* __restrict__ K, float* __restrict__ Q, float* __restrict__ V,
    int nodeOff)
{
  __shared__ float Xs[16 * HIDDEN];
  const int r0 = blockIdx.x * 16;
  for (int i = threadIdx.x; i < 16 * HIDDEN; i += 256)
    Xs[i] = X[(size_t)(r0 + (i >> 7)) * HIDDEN + (i & 127)];
  __syncthreads();

  const int wave = threadIdx.x >> 5;
  const int lane = threadIdx.x & 31;
  const int half = lane >> 4;      // 0: K pair {0,1}, 1: K pair {2,3}
  const int l16  = lane & 15;
  const int n0   = wave * 16;
  const int col  = n0 + l16;

  v8f ck = {0,0,0,0,0,0,0,0};
  v8f cq = {0,0,0,0,0,0,0,0};
  v8f cv = {0,0,0,0,0,0,0,0};

  for (int kb = 0; kb < HIDDEN; kb += 4) {
    const int k0 = kb + half * 2;
    v2f a;  a.x = Xs[l16 * HIDDEN + k0];        a.y = Xs[l16 * HIDDEN + k0 + 1];
    v2f fk; fk.x = Wk[(size_t)k0 * HIDDEN + col]; fk.y = Wk[(size_t)(k0+1) * HIDDEN + col];
    v2f fq; fq.x = Wq[(size_t)k0 * HIDDEN + col]; fq.y = Wq[(size_t)(k0+1) * HIDDEN + col];
    v2f fv; fv.x = Wv[(size_t)k0 * HIDDEN + col]; fv.y = Wv[(size_t)(k0+1) * HIDDEN + col];
    ck = wmma4(a, fk, ck);
    cq = wmma4(a, fq, cq);
    cv = wmma4(a, fv, cv);
  }

  const float bkc = bk[col], bqc = bq[col], bvc = bv[col];
#pragma unroll
  for (int i = 0; i < 8; ++i) {
    const int m = half * 8 + i;
    const size_t off = (size_t)(nodeOff + r0 + m) * HIDDEN + col;
    K[off] = ck[i] + bkc;
    Q[off] = cq[i] + bqc;
    V[off] = cv[i] + bvc;
  }
}

// ------------- per-relation head transforms: kr = k@a_rel, vr = v@m_rel ----
__global__ __launch_bounds__(256) void rel_kernel(
    const float* __restrict__ K, const float* __restrict__ V,
    const float* __restrict__ a_rel, const float* __restrict__ m_rel,
    float* __restrict__ KR, float* __restrict__ VR, int srcOff)
{
  const int r0   = blockIdx.x * 16;
  const int h    = threadIdx.x >> 5;   // wave == head
  const int lane = threadIdx.x & 31;
  const int half = lane >> 4;
  const int l16  = lane & 15;

  v8f ckr = {0,0,0,0,0,0,0,0};
  v8f cvr = {0,0,0,0,0,0,0,0};

  const size_t rowBase = (size_t)(srcOff + r0 + l16) * HIDDEN + h * HD;
  for (int kb = 0; kb < HD; kb += 4) {
    const int k0 = kb + half * 2;
    v2f ak; ak.x = K[rowBase + k0]; ak.y = K[rowBase + k0 + 1];
    v2f av; av.x = V[rowBase + k0]; av.y = V[rowBase + k0 + 1];
    v2f ba; ba.x = a_rel[h*256 + k0*16 + l16]; ba.y = a_rel[h*256 + (k0+1)*16 + l16];
    v2f bm; bm.x = m_rel[h*256 + k0*16 + l16]; bm.y = m_rel[h*256 + (k0+1)*16 + l16];
    ckr = wmma4(ak, ba, ckr);
    cvr = wmma4(av, bm, cvr);
  }
#pragma unroll
  for (int i = 0; i < 8; ++i) {
    const int m = half * 8 + i;
    const size_t off = (size_t)(r0 + m) * HIDDEN + h * HD + l16;
    KR[off] = ckr[i];
    VR[off] = cvr[i];
  }
}

// ---------------- softmax state init ----------------
__global__ void init_softmax(float* __restrict__ amax, float* __restrict__ ssum) {
  const int i = blockIdx.x * blockDim.x + threadIdx.x;
  if (i < NTOT * HEADS) { amax[i] = -INFINITY; ssum[i] = 0.0f; }
}
__global__ void zero_buf(float* __restrict__ p, int n) {
  const int i = blockIdx.x * blockDim.x + threadIdx.x;
  if (i < n) p[i] = 0.0f;
}

// ---------------- edge pass A: logits + segment max ----------------
__global__ __launch_bounds__(256) void edge_logits(
    const int* __restrict__ ei, const float* __restrict__ Q,
    const float* __restrict__ KR, const float* __restrict__ p_rel,
    float* __restrict__ logit, float* __restrict__ amax, int dstOff)
{
  const int idx = blockIdx.x * 256 + threadIdx.x;
  if (idx >= NEDGE * HEADS) return;
  const int e = idx >> 3, h = idx & 7;
  const int src = ei[e], dst = ei[NEDGE + e];
  const float4* qp = (const float4*)(Q  + (size_t)(dstOff + dst) * HIDDEN + h * HD);
  const float4* kp = (const float4*)(KR + (size_t)src * HIDDEN + h * HD);
  float s = 0.0f;
#pragma unroll
  for (int i = 0; i < 4; ++i) {
    const float4 qv = qp[i], kv = kp[i];
    s += qv.x*kv.x + qv.y*kv.y + qv.z*kv.z + qv.w*kv.w;
  }
  s *= p_rel[h] * ATT_SCALE;
  logit[idx] = s;
  atomicMaxF(amax + (size_t)(dstOff + dst) * HEADS + h, s);
}

// ---------------- edge pass B: exp + segment sum ----------------
__global__ __launch_bounds__(256) void edge_exp(
    const int* __restrict__ ei, float* __restrict__ logit,
    const float* __restrict__ amax, float* __restrict__ ssum, int dstOff)
{
  const int idx = blockIdx.x * 256 + threadIdx.x;
  if (idx >= NEDGE * HEADS) return;
  const int e = idx >> 3, h = idx & 7;
  const int dst = ei[NEDGE + e];
  const float v = expf(logit[idx] - amax[(size_t)(dstOff + dst) * HEADS + h]);
  logit[idx] = v;
  atomicAdd(ssum + (size_t)(dstOff + dst) * HEADS + h, v);
}

// ---------------- edge pass C: weighted scatter of messages ----------------
__global__ __launch_bounds__(256) void edge_agg(
    const int* __restrict__ ei, const float* __restrict__ logit,
    const float* __restrict__ ssum, const float* __restrict__ VR,
    float* __restrict__ agg, int dstOff)
{
  const int idx = blockIdx.x * 256 + threadIdx.x;
  if (idx >= NEDGE * HEADS) return;
  const int e = idx >> 3, h = idx & 7;
  const int src = ei[e], dst = ei[NEDGE + e];
  const float w = logit[idx] / (ssum[(size_t)(dstOff + dst) * HEADS + h] + 1e-16f);
  const float* vp = VR  + (size_t)src * HIDDEN + h * HD;
  float* ap       = agg + (size_t)(dstOff + dst) * HIDDEN + h * HD;
#pragma unroll
  for (int d = 0; d < HD; ++d) atomicAdd(ap + d, w * vp[d]);
}

// ------- output: gelu(agg)@Wo+bo, skip mix, residual, layernorm ------------
__global__ __launch_bounds__(256) void out_kernel(
    const float* __restrict__ AGG, const float* __restrict__ Xold,
    const float* __restrict__ Wo, const float* __restrict__ bo,
    const float* __restrict__ skip,
    const float* __restrict__ gamma, const float* __restrict__ betaLN,
    float* __restrict__ Xnew, int nodeOff)
{
  __shared__ float Gs[16 * HIDDEN];
  __shared__ float Os[16 * HIDDEN];
  const int r0 = blockIdx.x * 16;

  for (int i = threadIdx.x; i < 16 * HIDDEN; i += 256) {
    const float g = AGG[(size_t)(nodeOff + r0 + (i >> 7)) * HIDDEN + (i & 127)];
    Gs[i] = 0.5f * g * (1.0f + erff(g * 0.70710678118654752f)); // exact GELU
  }
  __syncthreads();

  const int wave = threadIdx.x >> 5;
  const int lane = threadIdx.x & 31;
  const int half = lane >> 4;
  const int l16  = lane & 15;
  const int col  = wave * 16 + l16;

  v8f c = {0,0,0,0,0,0,0,0};
  for (int kb = 0; kb < HIDDEN; kb += 4) {
    const int k0 = kb + half * 2;
    v2f a; a.x = Gs[l16 * HIDDEN + k0];         a.y = Gs[l16 * HIDDEN + k0 + 1];
    v2f b; b.x = Wo[(size_t)k0 * HIDDEN + col]; b.y = Wo[(size_t)(k0+1) * HIDDEN + col];
    c = wmma4(a, b, c);
  }
  const float boc = bo[col];
#pragma unroll
  for (int i = 0; i < 8; ++i)
    Os[(half * 8 + i) * HIDDEN + col] = c[i] + boc;
  __syncthreads();

  // layernorm( beta*o + (1-beta)*x + x ) row-wise; 1 wave handles 2 rows
  const float sb = 1.0f / (1.0f + expf(-skip[0]));
  for (int rr = wave; rr < 16; rr += 8) {
    const float* xrow = Xold + (size_t)(r0 + rr) * HIDDEN;
    float t[4];
    float s = 0.0f, s2 = 0.0f;
#pragma unroll
    for (int j = 0; j < 4; ++j) {
      const int cc = lane * 4 + j;
      const float o  = Os[rr * HIDDEN + cc];
      const float xo = xrow[cc];
      const float m  = sb * o + (1.0f - sb) * xo + xo;
      t[j] = m; s += m; s2 += m * m;
    }
    for (int off = 16; off >= 1; off >>= 1) {
      s  += __shfl_xor(s,  off, 32);
      s2 += __shfl_xor(s2, off, 32);
    }
    const float mean = s * (1.0f / 128.0f);
    const float var  = s2 * (1.0f / 128.0f) - mean * mean;
    const float inv  = rsqrtf(var + 1e-5f);
#pragma unroll
    for (int j = 0; j < 4; ++j) {
      const int cc = lane * 4 + j;
      Xnew[(size_t)(r0 + rr) * HIDDEN + cc] =
          (t[j] - mean) * inv * gamma[cc] + betaLN[cc];
    }
  }
}

// ============================ host launcher ================================
extern "C" void kernel_launch(void* const* d_in, const int* in_sizes, int n_in,
                              void* d_out, int out_size, void* d_ws, size_t ws_size,
                              hipStream_t stream)
{
  (void)n_in; (void)out_size; (void)ws_size;
  auto F = [&](int i) -> const float* { return (const float*)d_in[i]; };
  auto I = [&](int i) -> const int*   { return (const int*)d_in[i]; };

  // relations in EDGE_TYPES order: 0 d->d interacts, 1 d->p targets,
  // 2 p->d rev_targets, 3 p->p assoc
  const int NsrcA[4]  = {ND, ND, NP, NP};
  const int srcOff[4] = {0, 0, ND, ND};
  const int dstOff[4] = {0, ND, 0, ND};

  const float* xin[2];
  const int* ei[4];
  const float *Wk[2][2], *bk[2][2], *Wq[2][2], *bq[2][2], *Wv[2][2], *bv[2][2];
  const float *Wo[2][2], *bo[2][2], *skp[2][2], *gam[2][2], *bet[2][2];
  const float *Ar[2][4], *Mr[2][4], *Pr[2][4];

  const bool sorted = (in_sizes[0] == 2 * NEDGE); // jax pytree (sorted-key) order

  if (!sorted) {
    // dict-insertion order: x_dict, edge_index_dict, params
    xin[0] = F(0); xin[1] = F(1);
    ei[0] = I(2); ei[1] = I(3); ei[2] = I(4); ei[3] = I(5);
    for (int l = 0; l < 2; ++l) {
      const int L = 6 + 34 * l;
      for (int t = 0; t < 2; ++t) {
        const int b = L + 6 * t; // Wk,bk,Wq,bq,Wv,bv
        Wk[l][t]=F(b); bk[l][t]=F(b+1); Wq[l][t]=F(b+2); bq[l][t]=F(b+3);
        Wv[l][t]=F(b+4); bv[l][t]=F(b+5);
      }
      for (int r = 0; r < 4; ++r) { const int b = L + 12 + 3 * r;
        Ar[l][r]=F(b); Mr[l][r]=F(b+1); Pr[l][r]=F(b+2); }
      for (int t = 0; t < 2; ++t) { const int b = L + 24 + 3 * t;
        Wo[l][t]=F(b); bo[l][t]=F(b+1); skp[l][t]=F(b+2); }
      for (int t = 0; t < 2; ++t) {
        gam[l][t]=F(L+30+2*t); bet[l][t]=F(L+31+2*t); }   // gamma, beta
    }
  } else {
    // sorted keys: edge_index_dict, params, x_dict
    ei[0] = I(0); ei[1] = I(1); ei[3] = I(2); ei[2] = I(3); // assoc<rev_targets
    for (int l = 0; l < 2; ++l) {
      const int L = 4 + 34 * l;
      for (int t = 0; t < 2; ++t) {
        const int b = L + 6 * t; // Wk,Wq,Wv,bk,bq,bv (uppercase sorts first)
        Wk[l][t]=F(b); Wq[l][t]=F(b+1); Wv[l][t]=F(b+2);
        bk[l][t]=F(b+3); bq[l][t]=F(b+4); bv[l][t]=F(b+5);
      }
      for (int t = 0; t < 2; ++t) {
        bet[l][t]=F(L+12+2*t); gam[l][t]=F(L+13+2*t); }   // beta < gamma
      for (int t = 0; t < 2; ++t) { const int b = L + 16 + 3 * t;
        Wo[l][t]=F(b); bo[l][t]=F(b+1); skp[l][t]=F(b+2); } // Wo,bo,skip
      const int rmap[4] = {0, 1, 3, 2}; // interacts,targets,assoc,rev_targets
      for (int rr = 0; rr < 4; ++rr) { const int r = rmap[rr];
        const int b = L + 22 + 3 * rr;
        Ar[l][r]=F(b); Mr[l][r]=F(b+1); Pr[l][r]=F(b+2); }
    }
    xin[0] = F(72); xin[1] = F(73);
  }

  // ---------------- workspace carving (floats) ----------------
  float* ws = (float*)d_ws;
  size_t o = 0;
  float* Kb = ws + o; o += (size_t)NTOT * HIDDEN;
  float* Qb = ws + o; o += (size_t)NTOT * HIDDEN;
  float* Vb = ws + o; o += (size_t)NTOT * HIDDEN;
  float *KR[4], *VR[4];
  for (int r = 0; r < 4; ++r) {
    KR[r] = ws + o; o += (size_t)NsrcA[r] * HIDDEN;
    VR[r] = ws + o; o += (size_t)NsrcA[r] * HIDDEN;
  }
  float* logit[4];
  for (int r = 0; r < 4; ++r) { logit[r] = ws + o; o += (size_t)NEDGE * HEADS; }
  float* amax = ws + o; o += (size_t)NTOT * HEADS;
  float* ssum = ws + o; o += (size_t)NTOT * HEADS;
  float* agg  = ws + o; o += (size_t)NTOT * HIDDEN;
  float* x1[2];
  x1[0] = ws + o; o += (size_t)ND * HIDDEN;
  x1[1] = ws + o; o += (size_t)NP * HIDDEN;

  const int   Nt[2]   = {ND, NP};
  const int   tOff[2] = {0, ND};
  const float* xc[2]  = {xin[0], xin[1]};
  float* outBase = (float*)d_out;

  const int edgeBlocks = (NEDGE * HEADS + 255) / 256;

  for (int l = 0; l < 2; ++l) {
    for (int t = 0; t < 2; ++t)
      kqv_kernel<<<Nt[t] / 16, 256, 0, stream>>>(
          xc[t], Wk[l][t], bk[l][t], Wq[l][t], bq[l][t], Wv[l][t], bv[l][t],
          Kb, Qb, Vb, tOff[t]);

    init_softmax<<<(NTOT * HEADS + 255) / 256, 256, 0, stream>>>(amax, ssum);
    zero_buf<<<(NTOT * HIDDEN + 255) / 256, 256, 0, stream>>>(agg, NTOT * HIDDEN);

    for (int r = 0; r < 4; ++r)
      rel_kernel<<<NsrcA[r] / 16, 256, 0, stream>>>(
          Kb, Vb, Ar[l][r], Mr[l][r], KR[r], VR[r], srcOff[r]);

    for (int r = 0; r < 4; ++r)
      edge_logits<<<edgeBlocks, 256, 0, stream>>>(
          ei[r], Qb, KR[r], Pr[l][r], logit[r], amax, dstOff[r]);

    for (int r = 0; r < 4; ++r)
      edge_exp<<<edgeBlocks, 256, 0, stream>>>(
          ei[r], logit[r], amax, ssum, dstOff[r]);

    for (int r = 0; r < 4; ++r)
      edge_agg<<<edgeBlocks, 256, 0, stream>>>(
          ei[r], logit[r], ssum, VR[r], agg, dstOff[r]);

    float* xn[2];
    if (l == 0) { xn[0] = x1[0]; xn[1] = x1[1]; }
    else        { xn[0] = outBase; xn[1] = outBase + (size_t)ND * HIDDEN; }

    for (int t = 0; t < 2; ++t)
      out_kernel<<<Nt[t] / 16, 256, 0, stream>>>(
          agg, xc[t], Wo[l][t], bo[l][t], skp[l][t], gam[l][t], bet[l][t],
          xn[t], tOff[t]);

    xc[0] = xn[0]; xc[1] = xn[1];
  }
}